// SAGE_Encoder_38439957299938
// MI455X (gfx1250) — compile-verified
//
#include <hip/hip_runtime.h>
#include <hip/hip_bf16.h>
#include <cstddef>

typedef __attribute__((ext_vector_type(2))) float v2f;
typedef __attribute__((ext_vector_type(8))) float v8f;

#define FEAT 128

// ---------------- utility ----------------
__global__ void zero_f32(float* __restrict__ p, size_t n) {
  size_t i = (size_t)blockIdx.x * blockDim.x + threadIdx.x;
  size_t stride = (size_t)gridDim.x * blockDim.x;
  for (; i < n; i += stride) p[i] = 0.0f;
}

// ---------------- in-degree ----------------
__global__ void deg_kernel(const int* __restrict__ dst, float* __restrict__ deg, int E) {
  int e = blockIdx.x * blockDim.x + threadIdx.x;
  if (e < E) atomicAdd(deg + dst[e], 1.0f);
}

// ------------- edge scatter-add (wave per edge, float4 per lane) -------------
__global__ __launch_bounds__(256)
void scatter_edges(const float* __restrict__ xin, const int* __restrict__ src,
                   const int* __restrict__ dst, float* __restrict__ agg, int E) {
  int gid = blockIdx.x * blockDim.x + threadIdx.x;
  int e = gid >> 5;
  int lane = gid & 31;
  if (e >= E) return;
  int s = src[e], d = dst[e];
  const float4 v = *(const float4*)(xin + (size_t)s * FEAT + lane * 4);
  float* o = agg + (size_t)d * FEAT + lane * 4;
  atomicAdd(o + 0, v.x);
  atomicAdd(o + 1, v.y);
  atomicAdd(o + 2, v.z);
  atomicAdd(o + 3, v.w);
}

// ------------- fused SAGE GEMM: h = (agg/deg) @ Wl + bias + xin @ Wr -------------
// Each wave owns a 16-row stripe and ALL 128 output columns (8 accumulator
// tiles). A fragments are loaded once per k-step and reused across 8 WMMAs per
// operand matrix -> minimal A traffic; B reads hit W (64KB, L0/L2-resident).
// V_WMMA_F32_16X16X4_F32 keeps full f32 precision (GEMM is memory-bound).
__global__ __launch_bounds__(256)
void sage_gemm(const float* __restrict__ agg, const float* __restrict__ deg,
               const float* __restrict__ xin,
               const float* __restrict__ Wl, const float* __restrict__ bias,
               const float* __restrict__ Wr, float* __restrict__ hout, int N) {
  const int lane = threadIdx.x & 31;
  const int wave = threadIdx.x >> 5;
  const int r0   = (blockIdx.x * 8 + wave) << 4;  // 16 rows per wave
  const int half = lane >> 4;                     // lane-half selects K sub-pair
  const int lrow = lane & 15;

  // Clamp A-row for tail blocks: keeps EXEC all-ones for WMMA; stores guarded.
  const int arow = min(r0 + lrow, N - 1);
  const float invd = 1.0f / fmaxf(deg[arow], 1.0f);

  v8f c[8] = {v8f{}, v8f{}, v8f{}, v8f{}, v8f{}, v8f{}, v8f{}, v8f{}};

  for (int k = 0; k < FEAT; k += 4) {
    const int kb = k + half * 2;
    const float2 avl = *(const float2*)(agg + (size_t)arow * FEAT + kb);
    const float2 avr = *(const float2*)(xin + (size_t)arow * FEAT + kb);
    v2f a1, a2;
    a1.x = avl.x * invd;
    a1.y = avl.y * invd;
    a2.x = avr.x;
    a2.y = avr.y;
    const float* wl0 = Wl + (size_t)(kb + 0) * FEAT;
    const float* wl1 = Wl + (size_t)(kb + 1) * FEAT;
    const float* wr0 = Wr + (size_t)(kb + 0) * FEAT;
    const float* wr1 = Wr + (size_t)(kb + 1) * FEAT;
    #pragma unroll
    for (int t = 0; t < 8; ++t) {
      v2f b;
      b.x = wl0[t * 16 + lrow];
      b.y = wl1[t * 16 + lrow];
      c[t] = __builtin_amdgcn_wmma_f32_16x16x4_f32(false, a1, false, b, (short)0,
                                                   c[t], false, false);
      b.x = wr0[t * 16 + lrow];
      b.y = wr1[t * 16 + lrow];
      c[t] = __builtin_amdgcn_wmma_f32_16x16x4_f32(false, a2, false, b, (short)0,
                                                   c[t], false, false);
    }
  }

  // C/D layout: VGPR v, lane-half h -> row (v + 8h), col = t*16 + lrow
  if (r0 + 16 <= N) {
    // Fast path: whole 16-row stripe in bounds (always true when N % 16 == 0
    // and the wave isn't fully out of range) -> plain clause-able stores.
    #pragma unroll
    for (int t = 0; t < 8; ++t) {
      const float bv = bias[t * 16 + lrow];
      #pragma unroll
      for (int v = 0; v < 8; ++v) {
        hout[(size_t)(r0 + v + half * 8) * FEAT + t * 16 + lrow] = c[t][v] + bv;
      }
    }
  } else {
    // Cold tail: per-row predicated stores.
    #pragma unroll
    for (int t = 0; t < 8; ++t) {
      const float bv = bias[t * 16 + lrow];
      #pragma unroll
      for (int v = 0; v < 8; ++v) {
        const int row = r0 + v + half * 8;
        if (row < N) hout[(size_t)row * FEAT + t * 16 + lrow] = c[t][v] + bv;
      }
    }
  }
}

// ------------- BN stats: per-channel sum & sumsq -------------
__global__ __launch_bounds__(256)
void bn_stats(const float* __restrict__ h, float* __restrict__ stats, int N) {
  __shared__ float s1[256];
  __shared__ float s2[256];
  const int c = threadIdx.x & 127;
  const int ht = threadIdx.x >> 7;  // 2 row-groups per block
  float sum = 0.0f, sq = 0.0f;
  for (int r = blockIdx.x * 2 + ht; r < N; r += gridDim.x * 2) {
    float v = h[(size_t)r * FEAT + c];
    sum += v;
    sq += v * v;
  }
  s1[threadIdx.x] = sum;
  s2[threadIdx.x] = sq;
  __syncthreads();
  if (threadIdx.x < 128) {
    atomicAdd(stats + c,       s1[threadIdx.x] + s1[threadIdx.x + 128]);
    atomicAdd(stats + 128 + c, s2[threadIdx.x] + s2[threadIdx.x + 128]);
  }
}

// ------------- BN apply + ReLU (in place) -------------
__global__ __launch_bounds__(256)
void bn_apply(float* __restrict__ h, const float* __restrict__ stats,
              const float* __restrict__ gamma, const float* __restrict__ beta,
              int N, int total) {
  int i = blockIdx.x * blockDim.x + threadIdx.x;
  if (i >= total) return;
  const int c = i & 127;
  const float rn = 1.0f / (float)N;
  const float mu = stats[c] * rn;
  const float var = stats[128 + c] * rn - mu * mu;
  const float v = gamma[c] * (h[i] - mu) * rsqrtf(var + 1e-5f) + beta[c];
  h[i] = fmaxf(v, 0.0f);
}

// ------------- global mean pool: accumulate -------------
__global__ __launch_bounds__(256)
void pool_accum(const float* __restrict__ x, const int* __restrict__ batch,
                float* __restrict__ psum, float* __restrict__ cnt, int N) {
  int gid = blockIdx.x * blockDim.x + threadIdx.x;
  int n = gid >> 5;
  int lane = gid & 31;
  if (n >= N) return;
  const int g = batch[n];
  const float4 v = *(const float4*)(x + (size_t)n * FEAT + lane * 4);
  float* o = psum + (size_t)g * FEAT + lane * 4;
  atomicAdd(o + 0, v.x);
  atomicAdd(o + 1, v.y);
  atomicAdd(o + 2, v.z);
  atomicAdd(o + 3, v.w);
  if (lane == 0) atomicAdd(cnt + g, 1.0f);
}

__global__ void pool_final(const float* __restrict__ psum, const float* __restrict__ cnt,
                           float* __restrict__ out, int total) {
  int i = blockIdx.x * blockDim.x + threadIdx.x;
  if (i < total) out[i] = psum[i] / fmaxf(cnt[i >> 7], 1.0f);
}

extern "C" void kernel_launch(void* const* d_in, const int* in_sizes, int n_in,
                              void* d_out, int out_size, void* d_ws, size_t ws_size,
                              hipStream_t stream) {
  const float* x     = (const float*)d_in[0];
  const int*   ei    = (const int*)d_in[1];
  const int*   batch = (const int*)d_in[2];
  const float* W_l   = (const float*)d_in[3];
  const float* b_l   = (const float*)d_in[4];
  const float* W_r   = (const float*)d_in[5];
  const float* gamma = (const float*)d_in[6];
  const float* beta  = (const float*)d_in[7];
  float* out = (float*)d_out;

  const int N = in_sizes[0] / FEAT;
  const int E = in_sizes[1] / 2;
  const int* src = ei;
  const int* dst = ei + E;

  // workspace layout (floats)
  float* ws    = (float*)d_ws;
  const size_t ND = (size_t)N * FEAT;
  float* agg   = ws;              // N*128
  float* buf1  = agg + ND;        // N*128
  float* buf2  = buf1 + ND;       // N*128
  float* deg   = buf2 + ND;       // N
  float* stats = deg + N;         // 256 (sum | sumsq)
  float* psum  = stats + 256;     // G*128
  float* cnt   = psum + out_size; // G

  const int TB = 256;
  const int edgeWaveGrid = (E * 32 + TB - 1) / TB;
  const int gemmGrid = (N + 127) / 128;  // 128 rows per block (8 waves x 16)
  const int bnGrid = (N * FEAT + TB - 1) / TB;

  // degrees
  zero_f32<<<1024, TB, 0, stream>>>(deg, (size_t)N);
  deg_kernel<<<(E + TB - 1) / TB, TB, 0, stream>>>(dst, deg, E);

  const float* xin = x;
  float* bufs[2] = {buf1, buf2};
  for (int l = 0; l < 2; ++l) {
    float* hout = bufs[l];
    zero_f32<<<4096, TB, 0, stream>>>(agg, ND);
    zero_f32<<<1, TB, 0, stream>>>(stats, (size_t)256);
    scatter_edges<<<edgeWaveGrid, TB, 0, stream>>>(xin, src, dst, agg, E);
    sage_gemm<<<gemmGrid, TB, 0, stream>>>(agg, deg, xin,
                                           W_l + (size_t)l * FEAT * FEAT,
                                           b_l + (size_t)l * FEAT,
                                           W_r + (size_t)l * FEAT * FEAT,
                                           hout, N);
    bn_stats<<<1024, TB, 0, stream>>>(hout, stats, N);
    bn_apply<<<bnGrid, TB, 0, stream>>>(hout, stats,
                                        gamma + (size_t)l * FEAT,
                                        beta + (size_t)l * FEAT, N, N * FEAT);
    xin = hout;
  }

  // global mean pool
  zero_f32<<<64, TB, 0, stream>>>(psum, (size_t)out_size + 64);
  pool_accum<<<(N * 32 + TB - 1) / TB, TB, 0, stream>>>(buf2, batch, psum, cnt, N);
  pool_final<<<(out_size + TB - 1) / TB, TB, 0, stream>>>(psum, cnt, out, out_size);
}